// HyperTurbulentGenerator_87505663689240
// MI455X (gfx1250) — compile-verified
//
#include <hip/hip_runtime.h>
#include <hip/hip_bf16.h>

typedef __attribute__((ext_vector_type(2))) float v2f;
typedef __attribute__((ext_vector_type(8))) float v8f;

// ---------------------------------------------------------------------------
// Generic 64xN GEMM tile engine on LDS using V_WMMA_F32_16X16X4_F32.
//  C[MtilesxNtiles tiles of 16x16] = scale * (A @ B) (+ I)
//  A is MxK (optionally read transposed), B is KxN, all row-major f32 in LDS.
//  Caller supplies barriers. Uniform per-wave control flow (EXEC all-1s).
// Fragment layout per CDNA5 ISA 7.12.2:
//  A 16x4 : VGPR0 = K=0 (lanes 0-15) / K=2 (lanes 16-31), VGPR1 = K=1 / K=3
//  B 4x16 : VGPR0 = K=0 row (lanes=N) / K=2 row,          VGPR1 = K=1 / K=3
//  C 16x16: VGPR r = M=r (lanes 0-15) / M=r+8 (lanes 16-31), lane%16 = N
// ---------------------------------------------------------------------------
__device__ __forceinline__ void wmma_mm(const float* __restrict__ A, bool transA, int lda,
                                        const float* __restrict__ B, int ldb,
                                        float* __restrict__ C, int ldc,
                                        int Mtiles, int Ntiles, int K,
                                        float scale, bool addI)
{
    const int tid  = threadIdx.x;
    const int wave = tid >> 5;
    const int lane = tid & 31;
    const int half = lane >> 4;   // which 16-lane half
    const int l    = lane & 15;
    const int nw   = blockDim.x >> 5;
    const int ntiles = Mtiles * Ntiles;

    for (int tile = wave; tile < ntiles; tile += nw) {
        const int ti = tile / Ntiles;
        const int tj = tile % Ntiles;
        v8f acc = {0.f, 0.f, 0.f, 0.f, 0.f, 0.f, 0.f, 0.f};
        for (int kk = 0; kk < K; kk += 4) {
            v2f a, b;
            if (transA) {
                a.x = A[(kk + 2 * half + 0) * lda + ti * 16 + l];
                a.y = A[(kk + 2 * half + 1) * lda + ti * 16 + l];
            } else {
                a.x = A[(ti * 16 + l) * lda + kk + 2 * half + 0];
                a.y = A[(ti * 16 + l) * lda + kk + 2 * half + 1];
            }
            b.x = B[(kk + 2 * half + 0) * ldb + tj * 16 + l];
            b.y = B[(kk + 2 * half + 1) * ldb + tj * 16 + l];
            acc = __builtin_amdgcn_wmma_f32_16x16x4_f32(
                false, a, false, b, (short)0, acc, false, false);
        }
        const int col = tj * 16 + l;
#pragma unroll
        for (int r = 0; r < 8; ++r) {
            const int row = ti * 16 + 8 * half + r;
            float v = acc[r] * scale;
            if (addI && row == col) v += 1.0f;
            C[row * ldc + col] = v;
        }
    }
}

// ---------------------------------------------------------------------------
// Kernel 1: per batch b, build X = dt*(alpha*(K-K^T) - beta*R^T R),
// E = expm(X) via 6-term Taylor (Horner, 5 GEMMs), E16 = E^16 (4 squarings).
// One 4-wave block per batch; 3x 64x64 f32 LDS ping-pong buffers (48KB).
// ---------------------------------------------------------------------------
__global__ void __launch_bounds__(128)
build_expm_kernel(const float* __restrict__ kc, const float* __restrict__ rc,
                  const float* __restrict__ alpha_p, const float* __restrict__ beta_p,
                  const float* __restrict__ Kb, const float* __restrict__ Rb,
                  const float* __restrict__ ts,
                  float* __restrict__ Eout, float* __restrict__ E16out)
{
    __shared__ float lds0[4096];  // K_sum  -> P (Horner state)
    __shared__ float lds1[4096];  // R_sum  -> X
    __shared__ float lds2[4096];  // G      -> tmp
    const int b   = blockIdx.x;
    const int tid = threadIdx.x;

    const float dt = ts[0];
    const float al = alpha_p[b];
    const float be = beta_p[b];
    float kcoef[8], rcoef[8];
#pragma unroll
    for (int m = 0; m < 8; ++m) { kcoef[m] = kc[b * 8 + m]; rcoef[m] = rc[b * 8 + m]; }

    // K_sum / R_sum: bases are 256KB total -> L2 resident across all 512 blocks.
    for (int e = tid; e < 4096; e += 128) {
        float ks = 0.f, rs = 0.f;
#pragma unroll
        for (int m = 0; m < 8; ++m) {
            ks = fmaf(kcoef[m], Kb[m * 4096 + e], ks);
            rs = fmaf(rcoef[m], Rb[m * 4096 + e], rs);
        }
        lds0[e] = ks;
        lds1[e] = rs;
    }
    __syncthreads();

    // G = R_sum^T @ R_sum  (transposed-A read from LDS)
    wmma_mm(lds1, true, 64, lds1, 64, lds2, 64, 4, 4, 64, 1.0f, false);
    __syncthreads();

    // X = dt * (al*(Ks - Ks^T) - be*G)   into lds1 (R_sum dead)
    for (int e = tid; e < 4096; e += 128) {
        const int i = e >> 6, j = e & 63;
        lds1[e] = dt * (al * (lds0[e] - lds0[j * 64 + i]) - be * lds2[e]);
    }
    __syncthreads();

    // Horner: P = I + X/6, then P = I + (X@P)/k for k = 5..1  => E = P
    for (int e = tid; e < 4096; e += 128) {
        const int i = e >> 6, j = e & 63;
        lds0[e] = lds1[e] * (1.0f / 6.0f) + ((i == j) ? 1.0f : 0.0f);
    }
    __syncthreads();

    float* P = lds0;
    float* T = lds2;
    for (int k = 5; k >= 1; --k) {
        wmma_mm(lds1, false, 64, P, 64, T, 64, 4, 4, 64, 1.0f / (float)k, true);
        __syncthreads();
        float* t = P; P = T; T = t;
    }

    for (int e = tid; e < 4096; e += 128) Eout[(size_t)b * 4096 + e] = P[e];

    // E16 = E^16 via 4 squarings (ping-pong between lds2 and lds0; lds1 unused)
    float* Q = P;
    float* U = (P == lds0) ? lds2 : lds0;
    for (int it = 0; it < 4; ++it) {
        wmma_mm(Q, false, 64, Q, 64, U, 64, 4, 4, 64, 1.0f, false);
        __syncthreads();
        float* t = Q; Q = U; U = t;
    }
    for (int e = tid; e < 4096; e += 128) E16out[(size_t)b * 4096 + e] = Q[e];
}

// ---------------------------------------------------------------------------
// Kernel 2: per batch, z_{n+1} = E z_n.  First 16 steps as matvecs (128
// parallel 64-dot-products: row x {re,im}), forming Z block 64x32
// (cols 0-15 = re of steps s..s+15, cols 16-31 = im). Then 15 iterations of
// Z <- E16 @ Z via WMMA (4x2 C tiles across 4 waves), double-buffered, with
// coalesced float2 (re,im) output stores.
// ---------------------------------------------------------------------------
__global__ void __launch_bounds__(128)
propagate_kernel(const float* __restrict__ z0r, const float* __restrict__ z0i,
                 const float* __restrict__ E_all, const float* __restrict__ E16_all,
                 float2* __restrict__ out)
{
    __shared__ float Elds[4096];
    __shared__ float E16lds[4096];
    __shared__ float Zbuf[2][64 * 32];
    __shared__ float zr[64], zi[64];

    const int b   = blockIdx.x;
    const int tid = threadIdx.x;

    for (int e = tid; e < 4096; e += 128) {
        Elds[e]   = E_all[(size_t)b * 4096 + e];
        E16lds[e] = E16_all[(size_t)b * 4096 + e];
    }
    if (tid < 64) { zr[tid] = z0r[b * 64 + tid]; zi[tid] = z0i[b * 64 + tid]; }
    __syncthreads();

    const int row  = tid & 63;
    const int part = tid >> 6;  // 0 = real, 1 = imag

    float* Z = Zbuf[0];
    for (int s = 0; s < 16; ++s) {
        const float* v = part ? zi : zr;
        float acc = 0.f;
#pragma unroll 8
        for (int k2 = 0; k2 < 64; ++k2) acc = fmaf(Elds[row * 64 + k2], v[k2], acc);
        __syncthreads();
        if (part) zi[row] = acc; else zr[row] = acc;
        Z[row * 32 + part * 16 + s] = acc;
        __syncthreads();
    }

    // steps 1..16 -> output rows 0..15  (out[s] = E^{s+1} z0)
    for (int idx = tid; idx < 1024; idx += 128) {
        const int s = idx >> 6, d = idx & 63;
        out[((size_t)b * 256 + s) * 64 + d] = make_float2(Z[d * 32 + s], Z[d * 32 + 16 + s]);
    }

    float* Zc = Zbuf[0];
    float* Zn = Zbuf[1];
    for (int blk = 1; blk < 16; ++blk) {
        wmma_mm(E16lds, false, 64, Zc, 32, Zn, 32, 4, 2, 64, 1.0f, false);
        __syncthreads();
        for (int idx = tid; idx < 1024; idx += 128) {
            const int s = idx >> 6, d = idx & 63;
            out[((size_t)b * 256 + blk * 16 + s) * 64 + d] =
                make_float2(Zn[d * 32 + s], Zn[d * 32 + 16 + s]);
        }
        float* t = Zc; Zc = Zn; Zn = t;
    }
}

// ---------------------------------------------------------------------------
// Launch
// ---------------------------------------------------------------------------
extern "C" void kernel_launch(void* const* d_in, const int* in_sizes, int n_in,
                              void* d_out, int out_size, void* d_ws, size_t ws_size,
                              hipStream_t stream)
{
    const float* z0r = (const float*)d_in[0];  // (B,64)
    const float* z0i = (const float*)d_in[1];  // (B,64)
    const float* ts  = (const float*)d_in[2];  // (256,)
    const float* kc  = (const float*)d_in[3];  // (B,8)
    const float* rc  = (const float*)d_in[4];  // (B,8)
    const float* al  = (const float*)d_in[5];  // (B,1)
    const float* be  = (const float*)d_in[6];  // (B,1)
    const float* Kb  = (const float*)d_in[7];  // (8,64,64)
    const float* Rb  = (const float*)d_in[8];  // (8,64,64)

    const int B = in_sizes[0] / 64;  // 512

    float* E   = (float*)d_ws;                  // B * 4096 f32
    float* E16 = E + (size_t)B * 4096;          // B * 4096 f32 (16MB total)

    build_expm_kernel<<<B, 128, 0, stream>>>(kc, rc, al, be, Kb, Rb, ts, E, E16);
    propagate_kernel<<<B, 128, 0, stream>>>(z0r, z0i, E, E16, (float2*)d_out);
}